// Wspe_Wspa_Loss_15126874816815
// MI455X (gfx1250) — compile-verified
//
#include <hip/hip_runtime.h>
#include <hip/hip_bf16.h>

// Problem constants (hardcoded in the reference module)
#define GH 100
#define GW 100
#define GC 9
#define NPIX (GH * GW)

static __device__ __forceinline__ float inv_sqrt2() { return 0.70710678118654752440f; }
// 1 / (4/sqrt(2) + 4)
static __device__ __forceinline__ float inv_denom() { return 1.0f / 6.82842712474619009760f; }

typedef __attribute__((ext_vector_type(2))) float v2f;
typedef __attribute__((ext_vector_type(8))) float v8f;

// ---------------------------------------------------------------------------
// Kernel 0: zero the accumulators (ws[0..8] = per-channel sum of squares,
// out[0] = loss). Harness poisons but does not re-zero between replays.
// ---------------------------------------------------------------------------
__global__ void k_init(float* __restrict__ ws, float* __restrict__ out) {
    int t = threadIdx.x;
    if (t < GC) ws[t] = 0.0f;
    if (t == GC) out[0] = 0.0f;
}

// ---------------------------------------------------------------------------
// Kernel 1: per-channel sum of squares over the 10000 rows.
// Each thread owns whole rows (9 contiguous floats -> coalesced across the
// wave at 36B/row granularity, all L2-resident), keeps 9 register
// accumulators, shuffle-reduces within the wave, lane 0 atomically adds.
// ---------------------------------------------------------------------------
__global__ void k_colnorm(const float* __restrict__ abu, float* __restrict__ ws) {
    int tid    = blockIdx.x * blockDim.x + threadIdx.x;
    int stride = gridDim.x * blockDim.x;

    float acc[GC];
#pragma unroll
    for (int c = 0; c < GC; ++c) acc[c] = 0.0f;

    for (int r = tid; r < NPIX; r += stride) {
        const float* p = abu + r * GC;
#pragma unroll
        for (int c = 0; c < GC; ++c) {
            float v = p[c];
            acc[c] += v * v;
        }
    }

    // wave32 shuffle tree per channel
#pragma unroll
    for (int c = 0; c < GC; ++c) {
#pragma unroll
        for (int off = 16; off > 0; off >>= 1)
            acc[c] += __shfl_down(acc[c], off, 32);
    }
    if ((threadIdx.x & 31) == 0) {
#pragma unroll
        for (int c = 0; c < GC; ++c) atomicAdd(&ws[c], acc[c]);
    }
}

// ---------------------------------------------------------------------------
// Kernel 2: fused stencil + w_spe scaling + |.| + reduction to scalar.
// One thread per pixel; per-wave partials are collapsed with a single
// V_WMMA_F32_16X16X4_F32 against a ones B-matrix:
//   A[m][k]: lane m, VGPR0 holds the partial (K=0 for lanes 0-15, K=2 for
//   lanes 16-31), VGPR1 = 0.  With B = ones, D[m][n] = p_m + p_{m+16}.
//   Summing the 8 D VGPRs gives rows 0..7 (lanes 0-15) / rows 8..15
//   (lanes 16-31); one shfl_xor(16) yields the full 32-lane sum.
// EXEC is all-ones at the WMMA (guarded work rejoins before it).
// ---------------------------------------------------------------------------
__global__ void k_loss(const float* __restrict__ abu, const float* __restrict__ ws,
                       float* __restrict__ out) {
    int pix = blockIdx.x * blockDim.x + threadIdx.x;

    float partial = 0.0f;
    if (pix < NPIX) {
        int y = pix / GW;
        int x = pix - y * GW;

        // w_spe from the per-channel sum of squares (9 sqrts/thread: trivial)
        float wspe[GC];
#pragma unroll
        for (int c = 0; c < GC; ++c)
            wspe[c] = 1.0f / (sqrtf(ws[c]) + 1e-4f);

        float acc[GC];
#pragma unroll
        for (int c = 0; c < GC; ++c) acc[c] = 0.0f;

        // 8-neighbor weighted sum (zero padding at the borders)
#pragma unroll
        for (int dy = -1; dy <= 1; ++dy) {
#pragma unroll
            for (int dx = -1; dx <= 1; ++dx) {
                if (dx == 0 && dy == 0) continue;
                int ny = y + dy, nx = x + dx;
                if (ny < 0 || ny >= GH || nx < 0 || nx >= GW) continue;
                float w = (dx != 0 && dy != 0) ? inv_sqrt2() : 1.0f;
                const float* p = abu + (ny * GW + nx) * GC;
#pragma unroll
                for (int c = 0; c < GC; ++c) acc[c] += w * p[c];
            }
        }

        const float* pc = abu + pix * GC;
#pragma unroll
        for (int c = 0; c < GC; ++c)
            partial += fabsf(acc[c] * inv_denom() * wspe[c] * pc[c]);
    }

    // ---- WMMA wave reduction: D = A(partials) x B(ones) ----
    v2f a;
    a.x = partial;
    a.y = 0.0f;
    v2f b;
    b.x = 1.0f;
    b.y = 1.0f;
    v8f c = {};
    c = __builtin_amdgcn_wmma_f32_16x16x4_f32(
        /*neg_a=*/false, a, /*neg_b=*/false, b,
        /*c_mod=*/(short)0, c, /*reuse_a=*/false, /*reuse_b=*/false);

    float s = c[0] + c[1] + c[2] + c[3] + c[4] + c[5] + c[6] + c[7];
    float total = s + __shfl_xor(s, 16, 32);

    if ((threadIdx.x & 31) == 0)
        atomicAdd(out, total);
}

// ---------------------------------------------------------------------------
extern "C" void kernel_launch(void* const* d_in, const int* in_sizes, int n_in,
                              void* d_out, int out_size, void* d_ws, size_t ws_size,
                              hipStream_t stream) {
    (void)in_sizes; (void)n_in; (void)out_size; (void)ws_size;
    const float* abu = (const float*)d_in[0];
    float* out = (float*)d_out;
    float* ws  = (float*)d_ws;   // 9 floats: per-channel sum of squares

    k_init<<<1, 32, 0, stream>>>(ws, out);

    const int threads = 256;
    const int blocks  = (NPIX + threads - 1) / threads;  // 40 blocks

    k_colnorm<<<blocks, threads, 0, stream>>>(abu, ws);
    k_loss<<<blocks, threads, 0, stream>>>(abu, ws, out);
}